// AttnDecoderRNN_23252952940845
// MI455X (gfx1250) — compile-verified
//
#include <hip/hip_runtime.h>

#define HID 512
#define BSZ 32
#define SEQ 1024
#define VOC 50000
#define G4H 2048          // 4*H
#define K_OUT 1024        // 2H

typedef __attribute__((ext_vector_type(2))) float v2f;
typedef __attribute__((ext_vector_type(8))) float v8f;

static __device__ __forceinline__ float sigf(float x) { return 1.0f / (1.0f + expf(-x)); }

// ---------------------------------------------------------------------------
// Embedding gather + bias_dot[b] = we[b] . b_a   (one block per batch row)
// ---------------------------------------------------------------------------
__global__ void k_embed(const int* __restrict__ idx, const float* __restrict__ emb,
                        const float* __restrict__ b_a, float* __restrict__ we,
                        float* __restrict__ bias_dot) {
    __shared__ float red[HID];
    int b = blockIdx.x, t = threadIdx.x;
    long long row = (long long)idx[b] * HID;
    float v = emb[row + t];
    we[b * HID + t] = v;
    red[t] = v * b_a[t];
    __syncthreads();
    for (int off = HID / 2; off > 0; off >>= 1) {
        if (t < off) red[t] += red[t + off];
        __syncthreads();
    }
    if (t == 0) bias_dot[b] = red[0];
}

// ---------------------------------------------------------------------------
// LSTM gates via f32 WMMA:
//   gates[b,j] = sum_k X[b,k] * Wcat[j,k] + b_ih[j] + b_hh[j]
//   X = [out(512) | last_context(512) | h(512)], Wcat = [W_ih | W_hh]
// One wave per 16-column tile (j); M=32 as two 16x16 accumulators.
// K-loop split at region boundaries -> branch-free inner loops.
// ---------------------------------------------------------------------------
__global__ void k_gates_wmma(const float* __restrict__ xo, const float* __restrict__ xc,
                             const float* __restrict__ xh,
                             const float* __restrict__ Wih, const float* __restrict__ Whh,
                             const float* __restrict__ bih, const float* __restrict__ bhh,
                             float* __restrict__ gates) {
    int wave = threadIdx.x >> 5;
    int lane = threadIdx.x & 31;
    int tile = blockIdx.x * (blockDim.x >> 5) + wave;   // 0..127
    int j0 = tile * 16;
    int lm  = lane & 15;
    int kof = (lane >> 4) << 1;                          // lanes 16..31 hold K+2,K+3
    v8f d0 = {}; v8f d1 = {};

    const float* wih_row = Wih + (size_t)(j0 + lm) * (2 * HID) + kof;  // B row, cols kof..
    const float* whh_row = Whh + (size_t)(j0 + lm) * HID + kof;
    const float* a0_xo = xo + lm * HID + kof;
    const float* a1_xo = xo + (16 + lm) * HID + kof;
    const float* a0_xc = xc + lm * HID + kof;
    const float* a1_xc = xc + (16 + lm) * HID + kof;
    const float* a0_xh = xh + lm * HID + kof;
    const float* a1_xh = xh + (16 + lm) * HID + kof;

#pragma unroll 4
    for (int k = 0; k < HID; k += 4) {                   // X region: out, W_ih cols [0,512)
        v2f a0 = *(const v2f*)(a0_xo + k);
        v2f a1 = *(const v2f*)(a1_xo + k);
        v2f bb = *(const v2f*)(wih_row + k);
        d0 = __builtin_amdgcn_wmma_f32_16x16x4_f32(false, a0, false, bb, (short)0, d0, false, false);
        d1 = __builtin_amdgcn_wmma_f32_16x16x4_f32(false, a1, false, bb, (short)0, d1, false, false);
    }
#pragma unroll 4
    for (int k = 0; k < HID; k += 4) {                   // X region: last_context, W_ih cols [512,1024)
        v2f a0 = *(const v2f*)(a0_xc + k);
        v2f a1 = *(const v2f*)(a1_xc + k);
        v2f bb = *(const v2f*)(wih_row + HID + k);
        d0 = __builtin_amdgcn_wmma_f32_16x16x4_f32(false, a0, false, bb, (short)0, d0, false, false);
        d1 = __builtin_amdgcn_wmma_f32_16x16x4_f32(false, a1, false, bb, (short)0, d1, false, false);
    }
#pragma unroll 4
    for (int k = 0; k < HID; k += 4) {                   // X region: hidden, W_hh
        v2f a0 = *(const v2f*)(a0_xh + k);
        v2f a1 = *(const v2f*)(a1_xh + k);
        v2f bb = *(const v2f*)(whh_row + k);
        d0 = __builtin_amdgcn_wmma_f32_16x16x4_f32(false, a0, false, bb, (short)0, d0, false, false);
        d1 = __builtin_amdgcn_wmma_f32_16x16x4_f32(false, a1, false, bb, (short)0, d1, false, false);
    }

    int n  = lane & 15;
    int mh = (lane >> 4) << 3;                           // lanes 16..31 hold M+8
    int j  = j0 + n;
    float bias = bih[j] + bhh[j];
#pragma unroll
    for (int r = 0; r < 8; ++r) {
        gates[(size_t)(r + mh) * G4H + j]      = d0[r] + bias;
        gates[(size_t)(16 + r + mh) * G4H + j] = d1[r] + bias;
    }
}

// ---------------------------------------------------------------------------
// LSTM pointwise: gate order i,f,g,o
// ---------------------------------------------------------------------------
__global__ void k_lstm_pw(const float* __restrict__ gates, const float* __restrict__ c_prev,
                          float* __restrict__ h_out, float* __restrict__ c_out) {
    int t = blockIdx.x * blockDim.x + threadIdx.x;       // 0..B*H
    int b = t >> 9, hh = t & (HID - 1);
    const float* g = gates + (size_t)b * G4H;
    float i  = sigf(g[hh]);
    float f  = sigf(g[HID + hh]);
    float gg = tanhf(g[2 * HID + hh]);
    float o  = sigf(g[3 * HID + hh]);
    float c  = f * c_prev[t] + i * gg;
    c_out[t] = c;
    h_out[t] = o * tanhf(c);
}

// ---------------------------------------------------------------------------
// v[b,h] = sum_k we[b,k] * W_a[k,h]   (coalesced over h)
// ---------------------------------------------------------------------------
__global__ void k_attn_v(const float* __restrict__ we, const float* __restrict__ W_a,
                         float* __restrict__ v) {
    __shared__ float swe[HID];
    int b = blockIdx.x, h = threadIdx.x;
    swe[h] = we[b * HID + h];
    __syncthreads();
    float acc = 0.f;
    for (int k = 0; k < HID; ++k) acc += swe[k] * W_a[(size_t)k * HID + h];
    v[b * HID + h] = acc;
}

// ---------------------------------------------------------------------------
// scores[b,s] = enc[s,b,:] . v[b,:] + bias_dot[b]   (one wave per (s,b))
// ---------------------------------------------------------------------------
__global__ void k_scores(const float* __restrict__ enc, const float* __restrict__ v,
                         const float* __restrict__ bias_dot, float* __restrict__ attn) {
    int wave = threadIdx.x >> 5, lane = threadIdx.x & 31;
    int id = blockIdx.x * (blockDim.x >> 5) + wave;      // 0..S*B-1
    int s = id >> 5, b = id & 31;
    const float* e  = enc + ((size_t)s * BSZ + b) * HID;
    const float* vb = v + b * HID;
    float acc = 0.f;
#pragma unroll
    for (int t = 0; t < 16; ++t) { int i = lane + (t << 5); acc += e[i] * vb[i]; }
    for (int off = 16; off > 0; off >>= 1) acc += __shfl_xor(acc, off, 32);
    if (lane == 0) attn[b * SEQ + s] = acc + bias_dot[b];
}

// ---------------------------------------------------------------------------
// In-place softmax over S per batch row (written into d_out attn slot)
// ---------------------------------------------------------------------------
__global__ void k_softmax(float* __restrict__ attn) {
    __shared__ float red[256];
    int b = blockIdx.x, t = threadIdx.x;
    float* row = attn + b * SEQ;
    float x0 = row[t], x1 = row[t + 256], x2 = row[t + 512], x3 = row[t + 768];
    float m = fmaxf(fmaxf(x0, x1), fmaxf(x2, x3));
    red[t] = m; __syncthreads();
    for (int off = 128; off > 0; off >>= 1) { if (t < off) red[t] = fmaxf(red[t], red[t + off]); __syncthreads(); }
    m = red[0]; __syncthreads();
    x0 = expf(x0 - m); x1 = expf(x1 - m); x2 = expf(x2 - m); x3 = expf(x3 - m);
    red[t] = x0 + x1 + x2 + x3; __syncthreads();
    for (int off = 128; off > 0; off >>= 1) { if (t < off) red[t] += red[t + off]; __syncthreads(); }
    float inv = 1.0f / red[0];
    row[t] = x0 * inv; row[t + 256] = x1 * inv; row[t + 512] = x2 * inv; row[t + 768] = x3 * inv;
}

// ---------------------------------------------------------------------------
// context[b,h] = sum_s attn[b,s] * enc[s,b,h]   (enc hot in L2 from k_scores)
// ---------------------------------------------------------------------------
__global__ void k_context(const float* __restrict__ attn, const float* __restrict__ enc,
                          float* __restrict__ ctx) {
    __shared__ float sa[SEQ];
    int b = blockIdx.x, h = threadIdx.x;                 // 512 threads
    sa[h] = attn[b * SEQ + h];
    sa[h + HID] = attn[b * SEQ + h + HID];
    __syncthreads();
    float acc = 0.f;
    for (int s = 0; s < SEQ; ++s) acc += sa[s] * enc[((size_t)s * BSZ + b) * HID + h];
    ctx[b * HID + h] = acc;
}

// ---------------------------------------------------------------------------
// logits[b,v] = [h2 | context][b,:] . W_out[v,:] + b_out[v]  via f32 WMMA.
// W_out (204.8 MB) streamed with non-temporal loads to keep enc in L2.
// K-loop split at the h2/context boundary -> branch-free inner loops.
// ---------------------------------------------------------------------------
__global__ void k_logits_wmma(const float* __restrict__ xo, const float* __restrict__ xc,
                              const float* __restrict__ W_out, const float* __restrict__ b_out,
                              float* __restrict__ out) {
    int wave = threadIdx.x >> 5, lane = threadIdx.x & 31;
    int tile = blockIdx.x * (blockDim.x >> 5) + wave;
    if (tile >= VOC / 16) return;                        // wave-uniform guard
    size_t v0 = (size_t)tile * 16;
    int lm  = lane & 15;
    int kof = (lane >> 4) << 1;
    v8f d0 = {}; v8f d1 = {};

    const float* wrow  = W_out + (v0 + lm) * (size_t)K_OUT + kof;
    const float* a0_xo = xo + lm * HID + kof;
    const float* a1_xo = xo + (16 + lm) * HID + kof;
    const float* a0_xc = xc + lm * HID + kof;
    const float* a1_xc = xc + (16 + lm) * HID + kof;

#pragma unroll 4
    for (int k = 0; k < HID; k += 4) {                   // X region: h2
        v2f a0 = *(const v2f*)(a0_xo + k);
        v2f a1 = *(const v2f*)(a1_xo + k);
        v2f bb = __builtin_nontemporal_load((const v2f*)(wrow + k));
        d0 = __builtin_amdgcn_wmma_f32_16x16x4_f32(false, a0, false, bb, (short)0, d0, false, false);
        d1 = __builtin_amdgcn_wmma_f32_16x16x4_f32(false, a1, false, bb, (short)0, d1, false, false);
    }
#pragma unroll 4
    for (int k = 0; k < HID; k += 4) {                   // X region: context
        v2f a0 = *(const v2f*)(a0_xc + k);
        v2f a1 = *(const v2f*)(a1_xc + k);
        v2f bb = __builtin_nontemporal_load((const v2f*)(wrow + HID + k));
        d0 = __builtin_amdgcn_wmma_f32_16x16x4_f32(false, a0, false, bb, (short)0, d0, false, false);
        d1 = __builtin_amdgcn_wmma_f32_16x16x4_f32(false, a1, false, bb, (short)0, d1, false, false);
    }

    int n = lane & 15, mh = (lane >> 4) << 3;
    float bias = b_out[v0 + n];
#pragma unroll
    for (int r = 0; r < 8; ++r) {
        out[(size_t)(r + mh) * VOC + v0 + n]      = d0[r] + bias;
        out[(size_t)(16 + r + mh) * VOC + v0 + n] = d1[r] + bias;
    }
}

// ---------------------------------------------------------------------------
// In-place log_softmax over V per batch row
// ---------------------------------------------------------------------------
__global__ void k_logsoftmax(float* __restrict__ out) {
    __shared__ float red[256];
    int b = blockIdx.x, t = threadIdx.x;
    float* row = out + (size_t)b * VOC;
    float m = -3.402823466e38f;
    for (int i = t; i < VOC; i += 256) m = fmaxf(m, row[i]);
    red[t] = m; __syncthreads();
    for (int off = 128; off > 0; off >>= 1) { if (t < off) red[t] = fmaxf(red[t], red[t + off]); __syncthreads(); }
    m = red[0]; __syncthreads();
    float s = 0.f;
    for (int i = t; i < VOC; i += 256) s += expf(row[i] - m);
    red[t] = s; __syncthreads();
    for (int off = 128; off > 0; off >>= 1) { if (t < off) red[t] += red[t + off]; __syncthreads(); }
    float lse = m + logf(red[0]);
    for (int i = t; i < VOC; i += 256) row[i] = row[i] - lse;
}

extern "C" void kernel_launch(void* const* d_in, const int* in_sizes, int n_in,
                              void* d_out, int out_size, void* d_ws, size_t ws_size,
                              hipStream_t stream) {
    (void)in_sizes; (void)n_in; (void)out_size; (void)ws_size;
    const int*   word_input   = (const int*)d_in[0];
    const float* last_context = (const float*)d_in[1];
    const float* h0           = (const float*)d_in[2];
    const float* c0           = (const float*)d_in[3];
    const float* enc          = (const float*)d_in[4];
    const float* embedding    = (const float*)d_in[5];
    const float* W_ih         = (const float*)d_in[6];
    const float* W_hh         = (const float*)d_in[7];
    const float* b_ih         = (const float*)d_in[8];
    const float* b_hh         = (const float*)d_in[9];
    const float* W_a          = (const float*)d_in[10];
    const float* b_a          = (const float*)d_in[11];
    const float* W_out        = (const float*)d_in[12];
    const float* b_out        = (const float*)d_in[13];

    // d_out layout (flattened tuple): out[B,V], context[B,H], h[1,B,H], c[1,B,H], attn[1,B,S]
    float* out      = (float*)d_out;
    float* ctx_out  = out + (size_t)BSZ * VOC;
    float* h_out    = ctx_out + BSZ * HID;
    float* c_out    = h_out + BSZ * HID;
    float* attn_out = c_out + BSZ * HID;

    // workspace (~525 KB)
    float* ws       = (float*)d_ws;
    float* we       = ws;                       // B*H
    float* bias_dot = we + BSZ * HID;           // 64 (padded)
    float* gates    = bias_dot + 64;            // B*4H
    float* h1       = gates + BSZ * G4H;        // B*H
    float* c1       = h1 + BSZ * HID;           // B*H
    float* vv       = c1 + BSZ * HID;           // B*H

    k_embed<<<BSZ, HID, 0, stream>>>(word_input, embedding, b_a, we, bias_dot);

    // LSTM layer 1: x=[we | last_context], hidden=h0/c0
    k_gates_wmma<<<16, 256, 0, stream>>>(we, last_context, h0, W_ih, W_hh, b_ih, b_hh, gates);
    k_lstm_pw<<<(BSZ * HID) / 256, 256, 0, stream>>>(gates, c0, h1, c1);
    // LSTM layer 2 (same weights): x=[h1 | last_context], hidden=h1/c1
    k_gates_wmma<<<16, 256, 0, stream>>>(h1, last_context, h1, W_ih, W_hh, b_ih, b_hh, gates);
    k_lstm_pw<<<(BSZ * HID) / 256, 256, 0, stream>>>(gates, c1, h_out, c_out);

    // Attention (algebraically reduced: v = we @ W_a, scores = enc . v + we.b_a)
    k_attn_v<<<BSZ, HID, 0, stream>>>(we, W_a, vv);
    k_scores<<<(SEQ * BSZ) / 8, 256, 0, stream>>>(enc, vv, bias_dot, attn_out);
    k_softmax<<<BSZ, 256, 0, stream>>>(attn_out);
    k_context<<<BSZ, HID, 0, stream>>>(attn_out, enc, ctx_out);

    // Output projection + log_softmax (logits written straight into d_out)
    int ntiles = VOC / 16;                       // 3125
    k_logits_wmma<<<(ntiles + 7) / 8, 256, 0, stream>>>(h_out, ctx_out, W_out, b_out, out);
    k_logsoftmax<<<BSZ, 256, 0, stream>>>(out);
}